// ECTLayer_76948634075439
// MI455X (gfx1250) — compile-verified
//
#include <hip/hip_runtime.h>

// ECT layer: ect[b,s,t] = sum_{n: batch[n]==b} sigmoid(SCALE*(lin[s] - (x@v)[n,t]))
//
// Strategy (MI455X / gfx1250, wave32):
//  - nh = x@v via V_WMMA_F32_16X16X4_F32 (K=3 padded to 4), exact f32.
//  - sigmoid = v_rcp_f32(1 + v_exp_f32(log2e*SCALE*(nh - lin_s)))  [TRANS units]
//  - batch is sorted -> per-block LDS accumulator for the block's base graph,
//    in-register 8-node reduction before each ds_add_f32, single flush of
//    global_atomic_add_f32 at the end. Cross-graph stragglers -> direct atomics.

#define N_POINTS   65536
#define NDIMS      3
#define NUM_THETAS 64
#define BUMP_STEPS 32
#define NUM_GRAPHS 64
#define SCALE      100.0f
#define LOG2E      1.44269504088896340736f

#define THREADS          256
#define WAVES            (THREADS / 32)
#define NODES_PER_BLOCK  256
#define TILES_PER_WAVE   (NODES_PER_BLOCK / 16 / WAVES)   // = 2
#define OUT_PER_GRAPH    (BUMP_STEPS * NUM_THETAS)        // 2048

typedef float v2f __attribute__((ext_vector_type(2)));
typedef float v8f __attribute__((ext_vector_type(8)));

__global__ void ect_zero_kernel(float* __restrict__ out, int n) {
    int i = blockIdx.x * blockDim.x + threadIdx.x;
    if (i < n) out[i] = 0.0f;
}

__global__ __launch_bounds__(THREADS)
void ect_main_kernel(const float* __restrict__ x,
                     const float* __restrict__ v,
                     const float* __restrict__ lin,
                     const int*   __restrict__ batch,
                     float*       __restrict__ out) {
    __shared__ float q2[BUMP_STEPS];          // SCALE*LOG2E*lin[s]
    __shared__ float acc[OUT_PER_GRAPH];      // [s][t] partial sums, 8 KB

    const int tid  = threadIdx.x;
    const int lane = tid & 31;
    const int wave = tid >> 5;
    const int blockStart = blockIdx.x * NODES_PER_BLOCK;

    for (int i = tid; i < OUT_PER_GRAPH; i += THREADS) acc[i] = 0.0f;
    if (tid < BUMP_STEPS) q2[tid] = (SCALE * LOG2E) * lin[tid];
    __syncthreads();

    const int  b0 = batch[blockStart];        // block's majority graph (batch sorted)
    const int  n  = lane & 15;                // column index within 16-wide tile
    const bool hi = (lane >= 16);

    // B matrix (4x16 per theta-group), layout mirrors the documented 32-bit A
    // layout: lane = column N, VGPR0/1 = K pair, half-waves split K {0,1}|{2,3}.
    // K=3 row is the zero pad.
    v2f bmat[4];
#pragma unroll
    for (int g = 0; g < 4; ++g) {
        const int t = g * 16 + n;
        bmat[g].x = hi ? v[2 * NUM_THETAS + t] : v[0 * NUM_THETAS + t];
        bmat[g].y = hi ? 0.0f                  : v[1 * NUM_THETAS + t];
    }

    for (int tile = 0; tile < TILES_PER_WAVE; ++tile) {
        const int tileBase = blockStart + (wave * TILES_PER_WAVE + tile) * 16;

        // A matrix 16x4: lane = row M (=n), lo half holds K=0,1; hi half K=2,3(=0)
        v2f amat;
        {
            const float* xr = x + (size_t)(tileBase + n) * NDIMS;
            amat.x = hi ? xr[2] : xr[0];
            amat.y = hi ? 0.0f  : xr[1];
        }

        // This lane's 8 output rows are nodes rowBase..rowBase+7 (C/D layout:
        // VGPR r holds M = r + (hi?8:0)).
        const int rowBase = tileBase + (hi ? 8 : 0);
        int  bt[8];
        bool uniform = true;
#pragma unroll
        for (int r = 0; r < 8; ++r) {
            bt[r] = batch[rowBase + r];
            uniform = uniform && (bt[r] == b0);
        }

#pragma unroll
        for (int g = 0; g < 4; ++g) {
            v8f c = {};
            // D = A(16x4, f32) x B(4x16, f32) + 0  -> nh tile (16 nodes x 16 thetas)
            c = __builtin_amdgcn_wmma_f32_16x16x4_f32(
                    false, amat, false, bmat[g], (short)0, c, false, false);

            float pl[8];
#pragma unroll
            for (int r = 0; r < 8; ++r) pl[r] = (SCALE * LOG2E) * c[r];

            const int theta = g * 16 + n;
            float* accCol = &acc[theta];

            if (uniform) {
                for (int s = 0; s < BUMP_STEPS; ++s) {
                    const float qs = q2[s];
                    float sum = 0.0f;
#pragma unroll
                    for (int r = 0; r < 8; ++r) {
                        // sigmoid(SCALE*(lin-nh)) = 1/(1+exp2(log2e*SCALE*(nh-lin)))
                        const float e = __builtin_amdgcn_exp2f(pl[r] - qs);
                        sum += __builtin_amdgcn_rcpf(1.0f + e);
                    }
                    atomicAdd(accCol + s * NUM_THETAS, sum);   // ds_add_f32
                }
            } else {
                // Graph boundary inside this 8-node run (rare; batch sorted).
                for (int s = 0; s < BUMP_STEPS; ++s) {
                    const float qs = q2[s];
                    float sum = 0.0f;
#pragma unroll
                    for (int r = 0; r < 8; ++r) {
                        const float e   = __builtin_amdgcn_exp2f(pl[r] - qs);
                        const float val = __builtin_amdgcn_rcpf(1.0f + e);
                        if (bt[r] == b0) {
                            sum += val;
                        } else {
                            unsafeAtomicAdd(
                                &out[(size_t)bt[r] * OUT_PER_GRAPH + s * NUM_THETAS + theta],
                                val);
                        }
                    }
                    atomicAdd(accCol + s * NUM_THETAS, sum);
                }
            }
        }
    }

    __syncthreads();
    // Flush block partials for graph b0 (few blocks per graph -> light contention).
    const size_t outBase = (size_t)b0 * OUT_PER_GRAPH;
    for (int i = tid; i < OUT_PER_GRAPH; i += THREADS)
        unsafeAtomicAdd(&out[outBase + i], acc[i]);
}

extern "C" void kernel_launch(void* const* d_in, const int* in_sizes, int n_in,
                              void* d_out, int out_size, void* d_ws, size_t ws_size,
                              hipStream_t stream) {
    const float* x     = (const float*)d_in[0];   // [65536,3]
    const float* v     = (const float*)d_in[1];   // [3,64]
    const float* lin   = (const float*)d_in[2];   // [32]
    const int*   batch = (const int*)  d_in[3];   // [65536] sorted
    float*       out   = (float*)d_out;           // [64,32,64]

    (void)in_sizes; (void)n_in; (void)d_ws; (void)ws_size;

    ect_zero_kernel<<<(out_size + 255) / 256, 256, 0, stream>>>(out, out_size);
    ect_main_kernel<<<N_POINTS / NODES_PER_BLOCK, THREADS, 0, stream>>>(
        x, v, lin, batch, out);
}